// SelfAttention_4217657885103
// MI455X (gfx1250) — compile-verified
//
#include <hip/hip_runtime.h>

typedef float v2f __attribute__((ext_vector_type(2)));
typedef float v8f __attribute__((ext_vector_type(8)));

#define B_ 4
#define S_ 256
#define D_ 512
#define P_ 256
#define NEG_INF_ (-1e9f)

__device__ __forceinline__ float fast_tanh(float x) {
#if __has_builtin(__builtin_amdgcn_tanhf)
  return __builtin_amdgcn_tanhf(x);   // v_tanh_f32 if gfx1250 exposes it
#else
  // tanh(x) = sign(x) * (1 - e^{-2|x|}) / (1 + e^{-2|x|})   (stable for all x)
  float ax = __builtin_fabsf(x);
  float t  = __expf(-2.0f * ax);                         // v_exp_f32 fast path
  float r  = (1.0f - t) * __builtin_amdgcn_rcpf(1.0f + t); // v_rcp_f32
  return x < 0.0f ? -r : r;
#endif
}

// ---------------------------------------------------------------------------
// Kernel A: pk = seq @ W1^T, pq = seq @ W2^T  (fp32 WMMA, one wave per 16x16)
//   seq: [B*S, D] row-major, W: [P, D] row-major -> out: [B*S, P]
// A-frag (16x4 f32): lane<16 holds K={k0,k0+1}, lane>=16 holds K={k0+2,k0+3}
// B-frag (4x16 f32): symmetric; B[k][n] = W[n][k] so each lane's pair is a
//   contiguous float2 of W's row n.
// C/D: VGPR r, lane l -> row m0 + r + 8*(l>>4), col n0 + (l&15)
// ---------------------------------------------------------------------------
__global__ __launch_bounds__(256) void proj_wmma(
    const float* __restrict__ seq, const float* __restrict__ W1,
    const float* __restrict__ W2, float* __restrict__ pk,
    float* __restrict__ pq) {
  const int lane = threadIdx.x & 31;
  const int wid  = blockIdx.x * (blockDim.x >> 5) + (threadIdx.x >> 5);
  const int matsel = wid >> 10;        // 0 -> W1/pk, 1 -> W2/pq
  const int tile   = wid & 1023;       // 64 m-tiles x 16 n-tiles
  const int m0 = (tile >> 4) << 4;
  const int n0 = (tile & 15) << 4;
  const float* W   = matsel ? W2 : W1;
  float*       out = matsel ? pq : pk;

  const int hl  = lane >> 4;           // lane half selects K pair
  const int l16 = lane & 15;
  const int m = m0 + l16;              // A row held by this lane
  const int n = n0 + l16;              // B col held by this lane

  v8f acc = {};
  for (int k0 = 0; k0 < D_; k0 += 4) {
    const int k = k0 + (hl << 1);
    v2f a = *(const v2f*)(seq + m * D_ + k);   // seq[m][k..k+1]
    v2f b = *(const v2f*)(W   + n * D_ + k);   // W[n][k..k+1] == B[k..k+1][n]
    acc = __builtin_amdgcn_wmma_f32_16x16x4_f32(
        false, a, false, b, (short)0, acc, false, false);
  }
#pragma unroll
  for (int r = 0; r < 8; ++r)
    out[(m0 + r + (hl << 3)) * P_ + n0 + l16] = acc[r];
}

// ---------------------------------------------------------------------------
// Kernel B: per (b, query i): scores_j = v . tanh(pq_i + pk_j), masked softmax
// over keys, store weights (with query mask folded in).
// 256 threads = 8 waves; wave w handles keys j in [32w, 32w+32); lanes
// parallel over p (coalesced pk row loads) + wave32 shuffle reduction.
// ---------------------------------------------------------------------------
__global__ __launch_bounds__(256) void score_softmax(
    const float* __restrict__ pq, const float* __restrict__ pk,
    const float* __restrict__ v, const float* __restrict__ mask,
    float* __restrict__ wout) {
  const int bi  = blockIdx.x;          // b*256 + i
  const int b   = bi >> 8;
  const int tid = threadIdx.x;
  const int lane = tid & 31, wv = tid >> 5;

  __shared__ float pq_s[P_];
  __shared__ float v_s[P_];
  __shared__ float sc[S_];
  __shared__ float redm[8];
  __shared__ float reds[8];

  pq_s[tid] = pq[bi * P_ + tid];
  v_s[tid]  = v[tid];
  __syncthreads();

  const float* pkb = pk + b * S_ * P_;
  for (int jj = 0; jj < 32; ++jj) {
    const int j = (wv << 5) + jj;
    const float* row = pkb + j * P_;
    if (jj + 1 < 32)  // gfx1250 global_prefetch_b8 for next key row
      __builtin_prefetch(pkb + (j + 1) * P_ + lane * 8, 0, 1);
    float part = 0.0f;
#pragma unroll
    for (int q = 0; q < 8; ++q) {
      const int p = lane + (q << 5);
      part = __builtin_fmaf(v_s[p], fast_tanh(pq_s[p] + row[p]), part);
    }
#pragma unroll
    for (int off = 16; off > 0; off >>= 1)
      part += __shfl_xor(part, off, 32);
    if (lane == 0) sc[j] = part;
  }
  __syncthreads();

  // masked softmax over sc[0..255]; thread tid owns key j = tid
  const float km = mask[b * S_ + tid];
  float s = (km > 0.0f) ? sc[tid] : NEG_INF_;
  float mx = s;
#pragma unroll
  for (int off = 16; off > 0; off >>= 1)
    mx = fmaxf(mx, __shfl_xor(mx, off, 32));
  if (lane == 0) redm[wv] = mx;
  __syncthreads();
  float bm = redm[0];
#pragma unroll
  for (int w = 1; w < 8; ++w) bm = fmaxf(bm, redm[w]);

  float e  = __expf(s - bm);
  float sm = e;
#pragma unroll
  for (int off = 16; off > 0; off >>= 1)
    sm += __shfl_xor(sm, off, 32);
  if (lane == 0) reds[wv] = sm;
  __syncthreads();
  float tot = 0.0f;
#pragma unroll
  for (int w = 0; w < 8; ++w) tot += reds[w];

  const int   i  = bi & 255;
  const float qm = mask[b * S_ + i];   // query mask folded into weights
  wout[bi * S_ + tid] = (e / tot) * qm;
}

// ---------------------------------------------------------------------------
// Kernel C: attn_b [256x512] = weights_b [256x256] @ seq_b [256x512]
// (weights already carry the query mask, so attn is masked automatically)
// ---------------------------------------------------------------------------
__global__ __launch_bounds__(256) void attn_wmma(
    const float* __restrict__ weights, const float* __restrict__ seq,
    float* __restrict__ attn) {
  const int lane = threadIdx.x & 31;
  const int wid  = blockIdx.x * (blockDim.x >> 5) + (threadIdx.x >> 5);
  const int b    = wid >> 9;           // 4 batches x 512 tiles
  const int rest = wid & 511;          // 16 m-tiles x 32 n-tiles
  const int m0 = (rest >> 5) << 4;
  const int n0 = (rest & 31) << 4;
  const float* Wt = weights + b * S_ * S_;
  const float* Sq = seq     + b * S_ * D_;

  const int hl  = lane >> 4;
  const int l16 = lane & 15;
  const int m = m0 + l16;
  const int n = n0 + l16;

  v8f acc = {};
  for (int k0 = 0; k0 < S_; k0 += 4) {
    const int k = k0 + (hl << 1);
    v2f a = *(const v2f*)(Wt + m * S_ + k);    // weights[m][k..k+1]
    v2f bb;
    bb.x = Sq[k * D_ + n];                     // seq[k][n]
    bb.y = Sq[(k + 1) * D_ + n];               // seq[k+1][n]
    acc = __builtin_amdgcn_wmma_f32_16x16x4_f32(
        false, a, false, bb, (short)0, acc, false, false);
  }
  float* O = attn + b * S_ * D_;
#pragma unroll
  for (int r = 0; r < 8; ++r)
    O[(m0 + r + (hl << 3)) * D_ + n0 + l16] = acc[r];
}

extern "C" void kernel_launch(void* const* d_in, const int* in_sizes, int n_in,
                              void* d_out, int out_size, void* d_ws, size_t ws_size,
                              hipStream_t stream) {
  (void)in_sizes; (void)n_in; (void)out_size; (void)ws_size;
  const float* seq  = (const float*)d_in[0];
  const float* mask = (const float*)d_in[1];
  const float* W1   = (const float*)d_in[2];
  const float* W2   = (const float*)d_in[3];
  const float* v    = (const float*)d_in[4];

  float* attn = (float*)d_out;                 // [B,S,D]
  float* wout = attn + B_ * S_ * D_;           // [B,S,S]
  float* pk   = (float*)d_ws;                  // [B*S, P]
  float* pq   = pk + B_ * S_ * P_;             // [B*S, P]

  // 2048 waves for projections (1024 tiles x {pk,pq}), 8 waves/block
  proj_wmma<<<256, 256, 0, stream>>>(seq, W1, W2, pk, pq);
  // one block per (b, query i)
  score_softmax<<<B_ * S_, 256, 0, stream>>>(pq, pk, v, mask, wout);
  // 2048 waves for attention GEMM, 8 waves/block
  attn_wmma<<<256, 256, 0, stream>>>(wout, seq, attn);
}